// Net_86328842650410
// MI455X (gfx1250) — compile-verified
//
#include <hip/hip_runtime.h>
#include <math.h>

typedef float v2f __attribute__((ext_vector_type(2)));
typedef float v8f __attribute__((ext_vector_type(8)));

#define N_NODES 100000
#define N_EDGES 3200000
#define F_IN    1433
#define F_HID   16
#define F_OUT   7
#define CHUNK_K 512   // 512*16*4 = 32 KB LDS

// ---------------- degree / normalization ----------------
__global__ void k_deg_init(float* __restrict__ dinv) {
  int n = blockIdx.x * blockDim.x + threadIdx.x;
  if (n < N_NODES) dinv[n] = 1.0f;            // self-loop contributes 1
}

__global__ void k_deg_count(const int* __restrict__ row, float* __restrict__ dinv) {
  int e = blockIdx.x * blockDim.x + threadIdx.x;
  if (e < N_EDGES) atomicAdd(&dinv[row[e]], 1.0f);
}

__global__ void k_deg_rsqrt(float* __restrict__ dinv) {
  int n = blockIdx.x * blockDim.x + threadIdx.x;
  if (n < N_NODES) dinv[n] = rsqrtf(dinv[n]); // deg >= 1 always
}

// ---------------- layer 1 dense transform: h = x @ W1 (WMMA f32 16x16x4) ----
__global__ __launch_bounds__(256) void k_gemm1(const float* __restrict__ x,
                                               const float* __restrict__ W1,
                                               float* __restrict__ h) {
  __shared__ float ldsW[CHUNK_K * F_HID];

  const int lane = threadIdx.x & 31;
  const int wave = threadIdx.x >> 5;
  const int tile = blockIdx.x * 8 + wave;        // 16-row tile index
  const int col  = lane & 15;                    // N (and M for A loads)
  const int koff = (lane >> 4) << 1;             // 0 for lanes 0-15, 2 for 16-31
  const bool active = (tile * 16) < N_NODES;     // wave-uniform

  const int arow = tile * 16 + col;              // A: M = lane % 16
  const float* xrow = x + (size_t)(active ? arow : 0) * F_IN;

  v8f c = {0.f, 0.f, 0.f, 0.f, 0.f, 0.f, 0.f, 0.f};

  for (int kc = 0; kc < F_IN; kc += CHUNK_K) {
    int klen = F_IN - kc; if (klen > CHUNK_K) klen = CHUNK_K;

    // cooperative load of W1 chunk into LDS (row-major K x 16)
    for (int i = threadIdx.x; i < klen * F_HID; i += 256)
      ldsW[i] = W1[(size_t)kc * F_HID + i];
    __syncthreads();

    if (active) {
      const int kfull = klen & ~3;
      for (int k = 0; k < kfull; k += 4) {
        v2f a, b;
        a.x = xrow[kc + k + koff];
        a.y = xrow[kc + k + koff + 1];
        b.x = ldsW[(k + koff) * F_HID + col];
        b.y = ldsW[(k + koff + 1) * F_HID + col];
        c = __builtin_amdgcn_wmma_f32_16x16x4_f32(false, a, false, b,
                                                  (short)0, c, false, false);
      }
      if (kfull < klen) {                        // K remainder (1..3), zero-pad
        v2f a = {0.f, 0.f}, b = {0.f, 0.f};
        if (kfull + koff < klen) {
          a.x = xrow[kc + kfull + koff];
          b.x = ldsW[(kfull + koff) * F_HID + col];
        }
        if (kfull + koff + 1 < klen) {
          a.y = xrow[kc + kfull + koff + 1];
          b.y = ldsW[(kfull + koff + 1) * F_HID + col];
        }
        c = __builtin_amdgcn_wmma_f32_16x16x4_f32(false, a, false, b,
                                                  (short)0, c, false, false);
      }
    }
    __syncthreads();
  }

  if (active) {
    // D layout: lanes 0-15 -> M = v, lanes 16-31 -> M = v + 8; N = lane % 16
    const int mbase = tile * 16 + ((lane >> 4) << 3);
    #pragma unroll
    for (int v = 0; v < 8; ++v)
      h[(size_t)(mbase + v) * F_HID + col] = c[v];
  }
}

// ---------------- layer 1 edge aggregation (16 lanes per edge) ----------------
__global__ void k_agg1(const float* __restrict__ h, const int* __restrict__ row,
                       const int* __restrict__ col, const float* __restrict__ dinv,
                       float* __restrict__ acc) {
  long long tid = (long long)blockIdx.x * blockDim.x + threadIdx.x;
  int e = (int)(tid >> 4);
  int f = (int)(tid & 15);
  if (e < N_EDGES) {
    int r = row[e], cc = col[e];
    float nrm = dinv[r] * dinv[cc];
    atomicAdd(&acc[(size_t)r * F_HID + f], h[(size_t)cc * F_HID + f] * nrm);
  }
}

// self-loop term + bias + ReLU (in place on acc)
__global__ void k_fin1(const float* __restrict__ h, const float* __restrict__ dinv,
                       const float* __restrict__ b1, float* __restrict__ acc) {
  int t = blockIdx.x * blockDim.x + threadIdx.x;
  if (t < N_NODES * F_HID) {
    int n = t >> 4, f = t & 15;
    float d = dinv[n];
    float v = acc[t] + h[t] * d * d + b1[f];
    acc[t] = v > 0.f ? v : 0.f;
  }
}

// ---------------- layer 2 dense transform: h2 = relu_out @ W2 ----------------
__global__ void k_gemm2(const float* __restrict__ a, const float* __restrict__ W2,
                        float* __restrict__ h2) {
  int t = blockIdx.x * blockDim.x + threadIdx.x;
  if (t < N_NODES * F_OUT) {
    int n = t / F_OUT, j = t % F_OUT;
    const float* ar = a + (size_t)n * F_HID;
    float s = 0.f;
    #pragma unroll
    for (int k = 0; k < F_HID; ++k) s += ar[k] * W2[k * F_OUT + j];
    h2[t] = s;
  }
}

// ---------------- layer 2 edge aggregation (8 lanes per edge, f<7) -----------
__global__ void k_agg2(const float* __restrict__ h2, const int* __restrict__ row,
                       const int* __restrict__ col, const float* __restrict__ dinv,
                       float* __restrict__ acc) {
  long long tid = (long long)blockIdx.x * blockDim.x + threadIdx.x;
  int e = (int)(tid >> 3);
  int f = (int)(tid & 7);
  if (e < N_EDGES && f < F_OUT) {
    int r = row[e], cc = col[e];
    float nrm = dinv[r] * dinv[cc];
    atomicAdd(&acc[(size_t)r * F_OUT + f], h2[(size_t)cc * F_OUT + f] * nrm);
  }
}

// self term + bias + log_softmax -> out
__global__ void k_fin2(const float* __restrict__ h2, const float* __restrict__ dinv,
                       const float* __restrict__ b2, const float* __restrict__ acc,
                       float* __restrict__ out) {
  int n = blockIdx.x * blockDim.x + threadIdx.x;
  if (n < N_NODES) {
    float d = dinv[n], d2 = d * d;
    float v[F_OUT];
    float m = -INFINITY;
    #pragma unroll
    for (int j = 0; j < F_OUT; ++j) {
      v[j] = acc[n * F_OUT + j] + h2[n * F_OUT + j] * d2 + b2[j];
      m = fmaxf(m, v[j]);
    }
    float s = 0.f;
    #pragma unroll
    for (int j = 0; j < F_OUT; ++j) s += expf(v[j] - m);
    float ls = logf(s);
    #pragma unroll
    for (int j = 0; j < F_OUT; ++j) out[n * F_OUT + j] = v[j] - m - ls;
  }
}

extern "C" void kernel_launch(void* const* d_in, const int* in_sizes, int n_in,
                              void* d_out, int out_size, void* d_ws, size_t ws_size,
                              hipStream_t stream) {
  const float* x   = (const float*)d_in[0];
  const int*   row = (const int*)  d_in[1];
  const int*   col = (const int*)  d_in[2];
  const float* W1  = (const float*)d_in[3];
  const float* b1  = (const float*)d_in[4];
  const float* W2  = (const float*)d_in[5];
  const float* b2  = (const float*)d_in[6];
  float* out = (float*)d_out;

  // workspace layout (floats)
  float* ws   = (float*)d_ws;
  float* dinv = ws;                               // N
  float* h1   = dinv + N_NODES;                   // N*16
  float* acc1 = h1   + (size_t)N_NODES * F_HID;   // N*16
  float* h2   = acc1 + (size_t)N_NODES * F_HID;   // N*7
  float* acc2 = h2   + (size_t)N_NODES * F_OUT;   // N*7

  // zero accumulators every call (harness does not re-poison between replays)
  hipMemsetAsync(acc1, 0, (size_t)N_NODES * F_HID * sizeof(float), stream);
  hipMemsetAsync(acc2, 0, (size_t)N_NODES * F_OUT * sizeof(float), stream);

  // degrees -> dinv
  k_deg_init <<<(N_NODES + 255) / 256, 256, 0, stream>>>(dinv);
  k_deg_count<<<(N_EDGES + 255) / 256, 256, 0, stream>>>(row, dinv);
  k_deg_rsqrt<<<(N_NODES + 255) / 256, 256, 0, stream>>>(dinv);

  // layer 1
  const int n_tiles = (N_NODES + 15) / 16;                  // 6250
  k_gemm1<<<(n_tiles + 7) / 8, 256, 0, stream>>>(x, W1, h1);
  {
    long long work = (long long)N_EDGES * 16;
    k_agg1<<<(unsigned)((work + 255) / 256), 256, 0, stream>>>(h1, row, col, dinv, acc1);
  }
  k_fin1<<<(N_NODES * F_HID + 255) / 256, 256, 0, stream>>>(h1, dinv, b1, acc1);

  // layer 2
  k_gemm2<<<(N_NODES * F_OUT + 255) / 256, 256, 0, stream>>>(acc1, W2, h2);
  {
    long long work = (long long)N_EDGES * 8;
    k_agg2<<<(unsigned)((work + 255) / 256), 256, 0, stream>>>(h2, row, col, dinv, acc2);
  }
  k_fin2<<<(N_NODES + 255) / 256, 256, 0, stream>>>(h2, dinv, b2, acc2, out);
}